// ConditionalSlicedTransport_discrete_14774687498754
// MI455X (gfx1250) — compile-verified
//
#include <hip/hip_runtime.h>
#include <hip/hip_bf16.h>

#define NDIM   512
#define NCOMP  256
#define NCLASS 8
#define NBIN   100
#define NSAMP  131072

typedef __attribute__((ext_vector_type(2))) float v2f;
typedef __attribute__((ext_vector_type(8))) float v8f;
typedef __attribute__((ext_vector_type(4))) unsigned int u32x4;
typedef __attribute__((ext_vector_type(8))) int i32x8;
typedef __attribute__((ext_vector_type(4))) int i32x4;

#define AS1 __attribute__((address_space(1)))
#define AS3 __attribute__((address_space(3)))

// ---------------------------------------------------------------------------
// Kernel 1: data0 = data @ wT        (N x 512) @ (512 x 256) -> (N x 256)
// Block = 512 threads = 16 waves; block computes a 16-row x 256-col slab.
// The contiguous 32KB A slab is DMA'd into LDS by the Tensor Data Mover with
// hardware padding (2 DWORDs every 256 DWORDs -> 4-bank skew per 512-f32 row)
// so the 16 lanes reading 16 rows at the same K hit distinct banks.
// ---------------------------------------------------------------------------
__global__ __launch_bounds__(512)
void gemm_data0_kernel(const float* __restrict__ data,
                       const float* __restrict__ wT,
                       float* __restrict__ data0) {
    __shared__ float As[16 * 516];          // 8192 payload + 64 pad floats
    const int n0  = blockIdx.x * 16;
    const int tid = threadIdx.x;

    if (tid < 32) {                          // one wave issues the TDM DMA
        const unsigned lds_base =
            (unsigned)(unsigned long long)(AS3 char*)As;
        const unsigned long long ga =
            (unsigned long long)(const void*)(data + (size_t)n0 * NDIM);
        u32x4 g0;
        g0.x = 1u;                           // count=1, user descriptor
        g0.y = lds_base;                     // lds_addr [63:32]
        g0.z = (unsigned)(ga & 0xffffffffull);            // global_addr lo
        g0.w = (unsigned)((ga >> 32) & 0x01ffffffull)     // global_addr hi
             | (2u << 30);                                // type = 2 (image)
        i32x8 g1;
        g1[0] = (int)((2u << 16)             // data_size = 4 bytes
               |      (1u << 20)             // pad_enable
               |      (7u << 22)             // pad_interval: every 256 DWORDs
               |      (1u << 25));           // pad_amount: 2 DWORDs
        g1[1] = (int)(512u << 16);           // tensor_dim0 = 512 (lo16@[63:48])
        g1[2] = (int)(16u  << 16);           // tensor_dim1 = 16  (lo16@[95:80])
        g1[3] = (int)(512u << 16);           // tile_dim0   = 512 ([127:112])
        g1[4] = 16;                          // tile_dim1   = 16  ([143:128])
        g1[5] = 512;                         // tensor_dim0_stride = 512
        g1[6] = 0;
        g1[7] = 0;
        i32x4 gz4 = {0, 0, 0, 0};            // groups 2/3 unused (2-D tensor)
        i32x8 gz8 = {0, 0, 0, 0, 0, 0, 0, 0};
        __builtin_amdgcn_tensor_load_to_lds(g0, g1, gz4, gz4, gz8, 0);
        __builtin_amdgcn_s_wait_tensorcnt(0);
    }
    __syncthreads();

    const int wave = tid >> 5;               // 0..15 -> output col tile
    const int lane = tid & 31;
    const int half = lane >> 4;
    const int l    = lane & 15;
    const int j0   = wave * 16;

    v8f acc = {0.f, 0.f, 0.f, 0.f, 0.f, 0.f, 0.f, 0.f};

#pragma unroll 4
    for (int k = 0; k < NDIM; k += 4) {
        const int ka  = k + 2 * half;        // even, < 512 -> pair never pads
        const int pa  = l * 516 + ka + 2 * (ka >> 8);   // TDM-padded layout
        v2f a, b;
        a.x = As[pa];
        a.y = As[pa + 1];
        b.x = wT[(size_t)ka       * NCOMP + j0 + l];
        b.y = wT[(size_t)(ka + 1) * NCOMP + j0 + l];
        acc = __builtin_amdgcn_wmma_f32_16x16x4_f32(
            false, a, false, b, (short)0, acc, false, false);
    }

#pragma unroll
    for (int v = 0; v < 8; ++v) {
        const int row = n0 + v + 8 * half;   // C layout: M = v + 8*half
        data0[(size_t)row * NCOMP + j0 + l] = acc[v];
    }
}

// ---------------------------------------------------------------------------
// Kernel 2: per-element RQ spline with class routing; in-place delta = y - x;
// per-sample reduction of log|dy/dx| -> logj.
// ---------------------------------------------------------------------------
__global__ __launch_bounds__(NCOMP)
void spline_kernel(const int* __restrict__ label,
                   const float* __restrict__ kx,
                   const float* __restrict__ ky,
                   const float* __restrict__ kd,
                   float* __restrict__ buf,     // in: data0, out: delta
                   float* __restrict__ logj) {
    __shared__ float red[NCOMP];
    const int n = blockIdx.x;
    const int j = threadIdx.x;

    const float x    = buf[(size_t)n * NCOMP + j];
    const int   c    = label[n];
    const int   base = (c * NCOMP + j) * NBIN;

    // lower_bound == jnp.searchsorted(side='left'); knots hot in L2 (~2.4 MB)
    int lo = 0, hi = NBIN;
    while (lo < hi) {
        int mid = (lo + hi) >> 1;
        if (kx[base + mid] < x) lo = mid + 1; else hi = mid;
    }

    float y, ld;
    if (lo == 0) {
        const float d0 = kd[base];
        y  = ky[base] + d0 * (x - kx[base]);
        ld = logf(d0);
    } else if (lo == NBIN) {
        const float d1 = kd[base + NBIN - 1];
        y  = ky[base + NBIN - 1] + d1 * (x - kx[base + NBIN - 1]);
        ld = logf(d1);
    } else {
        const int kk = lo - 1;               // in [0, NBIN-2]
        const float x0 = kx[base + kk], x1 = kx[base + kk + 1];
        const float y0 = ky[base + kk], y1 = ky[base + kk + 1];
        const float d0 = kd[base + kk], d1 = kd[base + kk + 1];
        const float w  = x1 - x0;
        const float s  = (y1 - y0) / w;
        float xi = (x - x0) / w;
        xi = fminf(fmaxf(xi, 0.f), 1.f);
        const float xi1   = 1.f - xi;
        const float denom = s + (d0 + d1 - 2.f * s) * xi * xi1;
        y  = y0 + (y1 - y0) * (s * xi * xi + d0 * xi * xi1) / denom;
        ld = 2.f * logf(s)
           + logf(d1 * xi * xi + 2.f * s * xi * xi1 + d0 * xi1 * xi1)
           - 2.f * logf(denom);
    }

    buf[(size_t)n * NCOMP + j] = y - x;      // delta, in place

    red[j] = ld;
    __syncthreads();
#pragma unroll
    for (int s2 = NCOMP / 2; s2 > 0; s2 >>= 1) {
        if (j < s2) red[j] += red[j + s2];
        __syncthreads();
    }
    if (j == 0) logj[n] = red[0];
}

// ---------------------------------------------------------------------------
// Kernel 3: out = data + delta @ wT^T   (N x 256) @ (256 x 512) -> (N x 512)
// Block = 1024 threads = 32 waves; the 16x256 delta slab is staged in LDS via
// per-lane async global->LDS b128 copies (ASYNCcnt path), padded stride 260.
// ---------------------------------------------------------------------------
__global__ __launch_bounds__(1024)
void gemm_out_kernel(const float* __restrict__ data,
                     const float* __restrict__ wT,
                     const float* __restrict__ delta,
                     float* __restrict__ out) {
    __shared__ float Ds[16 * 260];
    const int n0  = blockIdx.x * 16;
    const int tid = threadIdx.x;

    {   // 16*256 floats = 1024 float4 -> exactly one async copy per lane
        const int flat = tid * 4;
        const int row  = flat >> 8;          // / 256
        const int col  = flat & 255;
        __builtin_amdgcn_global_load_async_to_lds_b128(
            (AS1 i32x4*)(delta + (size_t)n0 * NCOMP + flat),
            (AS3 i32x4*)&Ds[row * 260 + col],
            0, 0);
        __builtin_amdgcn_s_wait_asynccnt(0);
    }
    __syncthreads();

    const int wave = tid >> 5;               // 0..31 -> output col tile
    const int lane = tid & 31;
    const int half = lane >> 4;
    const int l    = lane & 15;
    const int j0   = wave * 16;

    v8f acc = {0.f, 0.f, 0.f, 0.f, 0.f, 0.f, 0.f, 0.f};

#pragma unroll 4
    for (int k = 0; k < NCOMP; k += 4) {
        const int ka = k + 2 * half;
        v2f a, b;
        a.x = Ds[l * 260 + ka];
        a.y = Ds[l * 260 + ka + 1];
        // B[k][col] = wT[col][k]: contiguous pair within one wT row
        b.x = wT[(size_t)(j0 + l) * NCOMP + ka];
        b.y = wT[(size_t)(j0 + l) * NCOMP + ka + 1];
        acc = __builtin_amdgcn_wmma_f32_16x16x4_f32(
            false, a, false, b, (short)0, acc, false, false);
    }

#pragma unroll
    for (int v = 0; v < 8; ++v) {
        const int row = n0 + v + 8 * half;
        const size_t o = (size_t)row * NDIM + j0 + l;
        out[o] = data[o] + acc[v];           // fused '+ data' epilogue
    }
}

// ---------------------------------------------------------------------------
extern "C" void kernel_launch(void* const* d_in, const int* in_sizes, int n_in,
                              void* d_out, int out_size, void* d_ws, size_t ws_size,
                              hipStream_t stream) {
    const float* data  = (const float*)d_in[0];
    const float* wT    = (const float*)d_in[1];
    const float* kx    = (const float*)d_in[2];
    const float* ky    = (const float*)d_in[3];
    const float* kd    = (const float*)d_in[4];
    const int*   label = (const int*)d_in[5];

    float* out  = (float*)d_out;                       // (N, 512)
    float* logj = out + (size_t)NSAMP * NDIM;          // (N,)
    float* buf  = (float*)d_ws;                        // (N, 256) data0 -> delta

    gemm_data0_kernel<<<NSAMP / 16, 512, 0, stream>>>(data, wT, buf);
    spline_kernel    <<<NSAMP, NCOMP, 0, stream>>>(label, kx, ky, kd, buf, logj);
    gemm_out_kernel  <<<NSAMP / 16, 1024, 0, stream>>>(data, wT, buf, out);
}